// GCN_83236466197087
// MI455X (gfx1250) — compile-verified
//
#include <hip/hip_runtime.h>

#define N_NODES 100000
#define N_EDGES 1600000
#define IN_DIM 128
#define HIDDEN 128
#define OUT_DIM 4

typedef __attribute__((ext_vector_type(2))) float v2f;
typedef __attribute__((ext_vector_type(8))) float v8f;

// ---------------------------------------------------------------------------
// Degree / normalization
// ---------------------------------------------------------------------------
__global__ void k_set_deg(float* __restrict__ deg) {
    int i = blockIdx.x * blockDim.x + threadIdx.x;
    if (i < N_NODES) deg[i] = 1.0f;  // self loop
}

__global__ void k_count_deg(const int* __restrict__ dst, float* __restrict__ deg) {
    int e = blockIdx.x * blockDim.x + threadIdx.x;
    if (e < N_EDGES) unsafeAtomicAdd(&deg[dst[e]], 1.0f);
}

__global__ void k_dinv(float* __restrict__ deg) {
    int i = blockIdx.x * blockDim.x + threadIdx.x;
    if (i < N_NODES) {
        float d = deg[i];
        deg[i] = (d > 0.0f) ? rsqrtf(d) : 0.0f;  // deg>=1 always, keep ref semantics
    }
}

// ---------------------------------------------------------------------------
// GEMM1: H1[100000,128] = X[100000,128] @ W1[128,128]   (fp32 WMMA 16x16x4)
// One wave computes one 16x16 tile. Block = 8 waves = one 16-row strip of
// nodes across all 128 output columns. 100000/16 = 6250 blocks exactly.
//
// A frag (16x4 f32, ISA 7.12.2): lanes 0-15 row M=lane hold K=0 (v0), K=1 (v1);
// lanes 16-31 hold K=2 (v0), K=3 (v1).  B frag (4x16): v0 = rows K=0/K=2 on
// half-waves, v1 = rows K=1/K=3; column N = lane%16.
// C/D (16x16 f32): VGPR r = row M=r (lanes 0-15) / M=r+8 (lanes 16-31).
// ---------------------------------------------------------------------------
__global__ void k_gemm1(const float* __restrict__ X, const float* __restrict__ W,
                        float* __restrict__ H1) {
    const int lane = threadIdx.x & 31;
    const int wave = threadIdx.x >> 5;
    const int half = lane >> 4;      // 0: K pair {0,1}, 1: K pair {2,3}
    const int l16  = lane & 15;
    const int row0 = blockIdx.x * 16;
    const int n0   = wave * 16;

    const float* Xrow = X + (long)(row0 + l16) * IN_DIM + 2 * half;
    const float* Wcol = W + (long)(2 * half) * HIDDEN + n0 + l16;

    v8f c = {};
    #pragma unroll 4
    for (int k = 0; k < IN_DIM; k += 4) {
        // A: two consecutive K values per lane (8B aligned)
        v2f a;
        {
            const float2 t = *(const float2*)(Xrow + k);
            a.x = t.x; a.y = t.y;
        }
        // B: rows k+2*half and k+2*half+1, column n0+l16 (coalesced over lanes)
        v2f b;
        b.x = Wcol[(long)k * HIDDEN];
        b.y = Wcol[(long)k * HIDDEN + HIDDEN];
        // Unconditional near-cache prefetch of the A stream. Past the row end
        // this touches the next tile's rows (still wanted); past the end of X
        // it's a speculative prefetch the hardware silently drops.
        __builtin_prefetch(Xrow + k + 16, 0, 3);
        c = __builtin_amdgcn_wmma_f32_16x16x4_f32(false, a, false, b,
                                                  (short)0, c, false, false);
    }

    float* Hout = H1 + (long)(row0 + 8 * half) * HIDDEN + n0 + l16;
    #pragma unroll
    for (int r = 0; r < 8; ++r) Hout[(long)r * HIDDEN] = c[r];
}

// ---------------------------------------------------------------------------
// Layer-1 accumulator init with fused self-loop + bias:
//   AGG[i,f] = dinv[i]^2 * H1[i,f] + b1[f]
// ---------------------------------------------------------------------------
__global__ void k_init_agg1(const float* __restrict__ H1, const float* __restrict__ dinv,
                            const float* __restrict__ b1, float* __restrict__ AGG) {
    long idx = (long)blockIdx.x * blockDim.x + threadIdx.x;
    if (idx < (long)N_NODES * HIDDEN) {
        int i = (int)(idx >> 7);   // /128
        int f = (int)(idx & 127);
        float di = dinv[i];
        AGG[idx] = di * di * H1[idx] + b1[f];
    }
}

// ---------------------------------------------------------------------------
// Layer-1 edge scatter: one wave per edge, lane l -> features [4l,4l+3].
//   AGG[dst, :] += dinv[src]*dinv[dst] * H1[src, :]
// Edge id is wave-uniform so src/dst/norm loads scalarize; the float4 gather
// is coalesced; 4 native f32 atomics per lane land in L2 (H1/AGG are 51 MB,
// L2 is 192 MB).
// ---------------------------------------------------------------------------
__global__ void k_agg1(const float* __restrict__ H1, const int* __restrict__ src,
                       const int* __restrict__ dst, const float* __restrict__ dinv,
                       float* __restrict__ AGG) {
    const int lane = threadIdx.x & 31;
    const int e = blockIdx.x * 8 + (threadIdx.x >> 5);
    if (e >= N_EDGES) return;
    const int s = src[e];
    const int d = dst[e];
    const float nrm = dinv[s] * dinv[d];
    const int f0 = lane * 4;
    const float4 v = *(const float4*)(H1 + (long)s * HIDDEN + f0);
    float* p = AGG + (long)d * HIDDEN + f0;
    unsafeAtomicAdd(p + 0, v.x * nrm);
    unsafeAtomicAdd(p + 1, v.y * nrm);
    unsafeAtomicAdd(p + 2, v.z * nrm);
    unsafeAtomicAdd(p + 3, v.w * nrm);
}

// ---------------------------------------------------------------------------
// GEMM2: M2[100000,4] = relu(H)[100000,128] @ W2[128,4]  (WMMA, N padded to 16)
// ReLU fused into the A-fragment load. One wave per 16-node tile.
// ---------------------------------------------------------------------------
__global__ void k_gemm2(const float* __restrict__ H, const float* __restrict__ W2,
                        float* __restrict__ M2) {
    const int lane = threadIdx.x & 31;
    const int wave = threadIdx.x >> 5;
    const int half = lane >> 4;
    const int l16  = lane & 15;
    const int tile = blockIdx.x * 8 + wave;
    const int row0 = tile * 16;
    if (row0 >= N_NODES) return;   // wave-uniform: active waves keep EXEC all-1s

    const float* Hrow = H + (long)(row0 + l16) * HIDDEN + 2 * half;
    const bool colOK = (l16 < OUT_DIM);

    v8f c = {};
    #pragma unroll 4
    for (int k = 0; k < HIDDEN; k += 4) {
        v2f a;
        {
            const float2 t = *(const float2*)(Hrow + k);
            a.x = fmaxf(t.x, 0.0f);   // fused ReLU
            a.y = fmaxf(t.y, 0.0f);
        }
        v2f b;
        const int kb = k + 2 * half;
        b.x = colOK ? W2[(long)kb * OUT_DIM + l16] : 0.0f;
        b.y = colOK ? W2[(long)(kb + 1) * OUT_DIM + l16] : 0.0f;
        __builtin_prefetch(Hrow + k + 16, 0, 3);
        c = __builtin_amdgcn_wmma_f32_16x16x4_f32(false, a, false, b,
                                                  (short)0, c, false, false);
    }

    if (colOK) {
        float* Mout = M2 + (long)(row0 + 8 * half) * OUT_DIM + l16;
        #pragma unroll
        for (int r = 0; r < 8; ++r) Mout[(long)r * OUT_DIM] = c[r];
    }
}

// ---------------------------------------------------------------------------
// Layer-2 accumulator init (self-loop + bias):  OUT[i,c] = dinv[i]^2*M2[i,c]+b2[c]
// ---------------------------------------------------------------------------
__global__ void k_init_out(const float* __restrict__ M2, const float* __restrict__ dinv,
                           const float* __restrict__ b2, float* __restrict__ OUT) {
    long idx = (long)blockIdx.x * blockDim.x + threadIdx.x;
    if (idx < (long)N_NODES * OUT_DIM) {
        int i = (int)(idx >> 2);
        int cidx = (int)(idx & 3);
        float di = dinv[i];
        OUT[idx] = di * di * M2[idx] + b2[cidx];
    }
}

// ---------------------------------------------------------------------------
// Layer-2 edge scatter on 4-wide messages: one thread per edge.
// ---------------------------------------------------------------------------
__global__ void k_agg2(const float* __restrict__ M2, const int* __restrict__ src,
                       const int* __restrict__ dst, const float* __restrict__ dinv,
                       float* __restrict__ OUT) {
    int e = blockIdx.x * blockDim.x + threadIdx.x;
    if (e >= N_EDGES) return;
    const int s = src[e];
    const int d = dst[e];
    const float nrm = dinv[s] * dinv[d];
    const float4 m = *(const float4*)(M2 + (long)s * OUT_DIM);
    float* p = OUT + (long)d * OUT_DIM;
    unsafeAtomicAdd(p + 0, m.x * nrm);
    unsafeAtomicAdd(p + 1, m.y * nrm);
    unsafeAtomicAdd(p + 2, m.z * nrm);
    unsafeAtomicAdd(p + 3, m.w * nrm);
}

// ---------------------------------------------------------------------------
extern "C" void kernel_launch(void* const* d_in, const int* in_sizes, int n_in,
                              void* d_out, int out_size, void* d_ws, size_t ws_size,
                              hipStream_t stream) {
    const float* X  = (const float*)d_in[0];
    const int*   EI = (const int*)d_in[1];   // [2, N_EDGES] int32
    const float* W1 = (const float*)d_in[2];
    const float* b1 = (const float*)d_in[3];
    const float* W2 = (const float*)d_in[4];
    const float* b2 = (const float*)d_in[5];
    const int* src = EI;
    const int* dst = EI + N_EDGES;
    float* OUT = (float*)d_out;

    float* H1   = (float*)d_ws;                       // 100000*128
    float* AGG  = H1 + (long)N_NODES * HIDDEN;        // 100000*128 (becomes h)
    float* M2   = AGG + (long)N_NODES * HIDDEN;       // 100000*4
    float* DINV = M2 + (long)N_NODES * OUT_DIM;       // 100000

    const int nodeBlocks = (N_NODES + 255) / 256;
    const int edgeBlocks = (N_EDGES + 255) / 256;

    // normalization: deg (with self loop) -> dinv = rsqrt(deg)
    k_set_deg<<<nodeBlocks, 256, 0, stream>>>(DINV);
    k_count_deg<<<edgeBlocks, 256, 0, stream>>>(dst, DINV);
    k_dinv<<<nodeBlocks, 256, 0, stream>>>(DINV);

    // layer 1
    k_gemm1<<<N_NODES / 16, 256, 0, stream>>>(X, W1, H1);
    k_init_agg1<<<(int)(((long)N_NODES * HIDDEN + 255) / 256), 256, 0, stream>>>(H1, DINV, b1, AGG);
    k_agg1<<<(N_EDGES + 7) / 8, 256, 0, stream>>>(H1, src, dst, DINV, AGG);

    // layer 2 (relu fused into GEMM2 A load; gemm before aggregation, as in ref)
    k_gemm2<<<(N_NODES / 16 + 7) / 8, 256, 0, stream>>>(AGG, W2, M2);
    k_init_out<<<(int)(((long)N_NODES * OUT_DIM + 255) / 256), 256, 0, stream>>>(M2, DINV, b2, OUT);
    k_agg2<<<edgeBlocks, 256, 0, stream>>>(M2, src, dst, DINV, OUT);
}